// TopKPooling_18159121728196
// MI455X (gfx1250) — compile-verified
//
#include <hip/hip_runtime.h>
#include <hip/hip_bf16.h>

// Problem constants (fixed shapes from the reference)
#define GB   64                    // graphs
#define NPG  2048                  // nodes per graph
#define NN   (GB * NPG)            // 131072 nodes
#define CC   128                   // channels
#define EE   (4 * 1024 * 1024)     // edges
#define KK   1024                  // kept per graph
#define BK   (GB * KK)             // 65536 kept nodes

// d_out layout (float elements), in reference return order
#define XOUT_OFF   0
#define EDGE_OFF   (BK * CC)              // 8388608
#define BATCH_OFF  (EDGE_OFF + 2 * EE)    // 16777216
#define PERM_OFF   (BATCH_OFF + BK)
#define SCORE_OFF  (PERM_OFF + BK)
#define VALID_OFF  (SCORE_OFF + BK)

#define TILE_PAD 132   // dwords per LDS tile row: 132 % 64 == 4 -> conflict-free b64 reads

typedef __attribute__((ext_vector_type(2))) float v2f;
typedef __attribute__((ext_vector_type(8))) float v8f;

#if __has_builtin(__builtin_amdgcn_tanhf)
#define TANHF(v) __builtin_amdgcn_tanhf(v)
#elif __has_builtin(__builtin_amdgcn_tanh_f32)
#define TANHF(v) __builtin_amdgcn_tanh_f32(v)
#else
#define TANHF(v) tanhf(v)
#endif

// ---------------------------------------------------------------------------
// K1: score[i] = tanh(dot(x[i], w) / ||w||)
// One wave computes 16 rows via V_WMMA_F32_16X16X4_F32; x tile staged to LDS
// with CDNA5 async global->LDS copies (TENSOR/ASYNC path, tracked by ASYNCcnt).
// ---------------------------------------------------------------------------
__global__ __launch_bounds__(128) void score_kernel(const float* __restrict__ x,
                                                    const float* __restrict__ w,
                                                    float* __restrict__ score) {
  __shared__ __attribute__((aligned(16))) float ws[CC];
  __shared__ __attribute__((aligned(16))) float tile[4 * 16 * TILE_PAD];

  const int t    = threadIdx.x;
  const int wave = t >> 5;
  const int lane = t & 31;

  ws[t] = w[t];                       // blockDim.x == CC == 128
  __syncthreads();

  const int rowBase = (blockIdx.x * 4 + wave) * 16;
  float* tw = &tile[wave * 16 * TILE_PAD];

  // ---- stage 16x128 f32 tile, one 512B row per async b128 instruction ----
#pragma unroll
  for (int r = 0; r < 16; ++r) {
    const float* gp = x + (size_t)(rowBase + r) * CC + lane * 4;
    unsigned ldsoff = (unsigned)(uintptr_t)(tw + r * TILE_PAD + lane * 4);
    asm volatile("global_load_async_to_lds_b128 %0, %1, off"
                 :: "v"(ldsoff), "v"(gp) : "memory");
  }
  asm volatile("s_wait_asynccnt 0" ::: "memory");

  // ---- 1/||w|| (wave-parallel reduce over LDS copy of w) ----
  float p0 = ws[lane], p1 = ws[lane + 32], p2 = ws[lane + 64], p3 = ws[lane + 96];
  float n2 = p0 * p0 + p1 * p1 + p2 * p2 + p3 * p3;
  for (int m = 16; m > 0; m >>= 1) n2 += __shfl_xor(n2, m, 32);
  const float inv = rsqrtf(n2);

  const int row  = lane & 15;          // A-matrix M index held by this lane
  const int koff = (lane >> 4) << 1;   // lane-half -> k sub-offset (same rule for A and B)

#if __has_builtin(__builtin_amdgcn_wmma_f32_16x16x4_f32)
  v8f acc = {};
#pragma unroll
  for (int kb = 0; kb < CC; kb += 4) {
    v2f a = *(const v2f*)(tw + row * TILE_PAD + kb + koff);  // x fragment (ds_load_b64)
    v2f b = *(const v2f*)(ws + kb + koff);                   // w broadcast fragment
    acc = __builtin_amdgcn_wmma_f32_16x16x4_f32(false, a, false, b,
                                                (short)0, acc, false, false);
  }
  // D layout: lanes 0-15 hold M=0..7 in acc[0..7]; lanes 16-31 hold M=8..15.
  if ((lane & 15) == 0) {
    const int mb = (lane >> 4) * 8;
#pragma unroll
    for (int r = 0; r < 8; ++r)
      score[rowBase + mb + r] = TANHF(acc[r] * inv);
  }
#else
  // VALU fallback: lanes l and l+16 split the K range of row (l&15)
  float s = 0.f;
#pragma unroll
  for (int k = 0; k < 64; ++k) {
    int kk = (lane >> 4) * 64 + k;
    s += tw[row * TILE_PAD + kk] * ws[kk];
  }
  s += __shfl_xor(s, 16, 32);
  if (lane < 16) score[rowBase + row] = TANHF(s * inv);
#endif
}

// ---------------------------------------------------------------------------
// K2: per-graph top-K via in-LDS bitonic sort of 2048 64-bit keys.
// key = (~sortable(score) << 32) | local_idx  -> ascending sort == descending
// score with ascending-index tie-break (matches jax.lax.top_k).
// ---------------------------------------------------------------------------
__global__ __launch_bounds__(1024) void topk_kernel(const float* __restrict__ score,
                                                    int* __restrict__ perm,
                                                    int* __restrict__ mask,
                                                    float* __restrict__ out) {
  __shared__ unsigned long long keys[NPG];
  const int t    = threadIdx.x;
  const int g    = blockIdx.x;
  const int base = g * NPG;

  // relabel map init: this graph owns mask[base .. base+NPG)
  mask[base + t] = -1;
  mask[base + t + 1024] = -1;

#pragma unroll
  for (int j = 0; j < 2; ++j) {
    int i = t + j * 1024;
    unsigned b = __float_as_uint(score[base + i]);
    unsigned u = (b & 0x80000000u) ? ~b : (b | 0x80000000u);   // order-preserving map
    keys[i] = ((unsigned long long)(~u) << 32) | (unsigned)i;
  }
  __syncthreads();

  for (int size = 2; size <= NPG; size <<= 1) {
    for (int stride = size >> 1; stride > 0; stride >>= 1) {
      int i = ((t & ~(stride - 1)) << 1) | (t & (stride - 1));
      int j = i | stride;
      unsigned long long a = keys[i], b = keys[j];
      bool up = ((i & size) == 0);
      if ((a > b) == up) { keys[i] = b; keys[j] = a; }
      __syncthreads();
    }
  }

  // first K ascending keys == top-K
  unsigned long long kv = keys[t];
  int li = (int)(unsigned)(kv & 0xFFFFFFFFull);
  unsigned u = ~(unsigned)(kv >> 32);
  unsigned b = (u & 0x80000000u) ? (u & 0x7FFFFFFFu) : ~u;     // recover exact score bits
  float s = __uint_as_float(b);

  int node = base + li;
  int nid  = g * KK + t;
  perm[nid]  = node;
  mask[node] = nid;
  out[BATCH_OFF + nid] = (float)g;
  out[PERM_OFF  + nid] = (float)node;
  out[SCORE_OFF + nid] = s;
}

// ---------------------------------------------------------------------------
// K3: x_out[r] = x[perm[r]] * score[perm[r]] ; wave per row, float4 per lane
// (one 512B fully-coalesced transaction per row read and per row write).
// ---------------------------------------------------------------------------
__global__ __launch_bounds__(256) void gather_kernel(const float* __restrict__ x,
                                                     const float* __restrict__ score,
                                                     const int* __restrict__ perm,
                                                     float* __restrict__ out) {
  const int row  = (int)((blockIdx.x * blockDim.x + threadIdx.x) >> 5);
  const int lane = threadIdx.x & 31;
  const int node = perm[row];
  const float s  = score[node];
  const float4 v = *(const float4*)(x + (size_t)node * CC + lane * 4);
  float4 o;
  o.x = v.x * s; o.y = v.y * s; o.z = v.z * s; o.w = v.w * s;
  *(float4*)(out + (size_t)row * CC + lane * 4) = o;
}

// ---------------------------------------------------------------------------
// K4: edge relabel + validity mask.
// ---------------------------------------------------------------------------
__global__ __launch_bounds__(256) void edge_kernel(const int* __restrict__ ei,
                                                   const int* __restrict__ mask,
                                                   float* __restrict__ out) {
  const int e = blockIdx.x * blockDim.x + threadIdx.x;
  const int r = mask[ei[e]];
  const int c = mask[ei[e + EE]];
  const bool val = (r >= 0) && (c >= 0);
  out[EDGE_OFF + e]      = (float)(val ? r : -1);
  out[EDGE_OFF + EE + e] = (float)(val ? c : -1);
  out[VALID_OFF + e]     = val ? 1.0f : 0.0f;
}

extern "C" void kernel_launch(void* const* d_in, const int* in_sizes, int n_in,
                              void* d_out, int out_size, void* d_ws, size_t ws_size,
                              hipStream_t stream) {
  const float* x    = (const float*)d_in[0];   // [N, C] f32
  const int*   eidx = (const int*)d_in[1];     // [2, E] i32
  // d_in[2] = batch (implicit: g = node / NPG)
  const float* w    = (const float*)d_in[3];   // [1, C] f32
  float* out = (float*)d_out;

  // workspace: score[N] f32 | perm[BK] i32 | mask[N] i32  (~1.25 MB)
  float* scoreF = (float*)d_ws;
  int*   permI  = (int*)((char*)d_ws + (size_t)NN * 4);
  int*   maskI  = (int*)((char*)d_ws + (size_t)NN * 4 + (size_t)BK * 4);

  score_kernel <<<NN / 64, 128, 0, stream>>>(x, w, scoreF);          // 4 waves x 16 rows
  topk_kernel  <<<GB, 1024, 0, stream>>>(scoreF, permI, maskI, out); // 1 WG per graph
  gather_kernel<<<BK / 8, 256, 0, stream>>>(x, scoreF, permI, out);  // wave per row
  edge_kernel  <<<EE / 256, 256, 0, stream>>>(eidx, maskI, out);
}